// GATEncoder_65171833749591
// MI455X (gfx1250) — compile-verified
//
#include <hip/hip_runtime.h>
#include <hip/hip_bf16.h>

typedef __attribute__((ext_vector_type(2))) float v2f;
typedef __attribute__((ext_vector_type(8))) float v8f;

#define CH 64
#define NEG_SLOPE 0.2f

// ---------------------------------------------------------------------------
// GEMM: Y[N,C] = act(X[N,K]) @ W[K,C] + bias, via V_WMMA_F32_16X16X4_F32.
// One wave computes one 16x16 output tile; loops K in steps of 4.
// A layout (16x4 f32): lane<16 -> row=lane,  a.x=K0,a.y=K1 ; lane>=16 -> K2,K3
// B layout (4x16 f32): lane<16 -> col=lane,  b.x=K0,b.y=K1 ; lane>=16 -> K2,K3
// C/D layout: VGPR r: lanes0-15 -> row r, lanes16-31 -> row r+8.
// RELU is a compile-time template parameter so the no-activation instance
// has a pure load->wmma inner loop (no per-element cndmask).
// ---------------------------------------------------------------------------
template <int RELU>
__global__ __launch_bounds__(256) void gemm_wmma_f32(
    const float* __restrict__ X, const float* __restrict__ W,
    const float* __restrict__ bias, float* __restrict__ Y,
    int N, int K)
{
    const int wavesPerBlock = blockDim.x >> 5;
    const int wid  = blockIdx.x * wavesPerBlock + (threadIdx.x >> 5);
    const int tilesN = CH >> 4;                       // 4 col tiles
    const int tm = wid / tilesN;
    const int tn = wid - tm * tilesN;
    if (tm * 16 >= N) return;                         // wave-uniform

    const int lane = threadIdx.x & 31;
    const int half = lane >> 4;                       // 0 or 1
    const int l16  = lane & 15;
    const int m0 = tm * 16, n0 = tn * 16;

    const float* xrow = X + (size_t)(m0 + l16) * K;   // A: row per lane
    const float* wcol = W + (n0 + l16);               // B: col per lane

    v8f acc = {};
    for (int k = 0; k < K; k += 4) {
        const int ka = k + half * 2;
        v2f a, b;
        a.x = xrow[ka];
        a.y = xrow[ka + 1];
        if (RELU) { a.x = fmaxf(a.x, 0.f); a.y = fmaxf(a.y, 0.f); }
        b.x = wcol[(size_t)ka * CH];
        b.y = wcol[(size_t)(ka + 1) * CH];
        acc = __builtin_amdgcn_wmma_f32_16x16x4_f32(
            /*neg_a=*/false, a, /*neg_b=*/false, b,
            /*c_mod=*/(short)0, acc, /*reuse_a=*/false, /*reuse_b=*/false);
    }

    const float bn = bias[n0 + l16];
    float* yout = Y + (size_t)(m0 + half * 8) * CH + n0 + l16;
#pragma unroll
    for (int r = 0; r < 8; ++r)
        yout[(size_t)r * CH] = acc[r] + bn;
}

// Order-preserving float <-> uint key for atomic max on f32 values.
__device__ __forceinline__ unsigned fkey(float f) {
    unsigned u = __float_as_uint(f);
    return (u & 0x80000000u) ? ~u : (u | 0x80000000u);
}
__device__ __forceinline__ float funkey(unsigned k) {
    return (k & 0x80000000u) ? __uint_as_float(k & 0x7FFFFFFFu)
                             : __uint_as_float(~k);
}

// Init: out rows = bias broadcast; per-node max-key = 0 (== -inf), denom = 0.
__global__ void init_layer(float* __restrict__ out, const float* __restrict__ bias,
                           unsigned* __restrict__ lmax, float* __restrict__ denom,
                           int N)
{
    const int i = blockIdx.x * blockDim.x + threadIdx.x;
    if (i < N * CH) out[i] = bias[i & (CH - 1)];
    if (i < N) { lmax[i] = 0u; denom[i] = 0.f; }
}

// One wave per edge: m = leaky(xl[src]+xr[dst]+ea*We); logit = m.att;
// store logit, atomic-max into per-dst key. Lane i handles channels 2i,2i+1
// -> each row read is one coalesced 256B transaction (L2-resident).
__global__ __launch_bounds__(256) void edge_logits(
    const long long* __restrict__ ei, const float* __restrict__ ea,
    const float* __restrict__ xl, const float* __restrict__ xr,
    const float* __restrict__ We, const float* __restrict__ att,
    float* __restrict__ logits, unsigned* __restrict__ lmax, int E)
{
    const int e = blockIdx.x * (blockDim.x >> 5) + (threadIdx.x >> 5);
    if (e >= E) return;                               // wave-uniform
    const int lane = threadIdx.x & 31;

    const int s = (int)ei[e];
    const int d = (int)ei[(size_t)E + e];
    const float eav = ea[e];

    const v2f* xl2 = (const v2f*)(xl + (size_t)s * CH);
    const v2f* xr2 = (const v2f*)(xr + (size_t)d * CH);
    const v2f* we2 = (const v2f*)We;
    const v2f* at2 = (const v2f*)att;

    v2f m = xl2[lane] + xr2[lane] + eav * we2[lane];
    m.x = m.x > 0.f ? m.x : NEG_SLOPE * m.x;
    m.y = m.y > 0.f ? m.y : NEG_SLOPE * m.y;
    const v2f a = at2[lane];
    float p = m.x * a.x + m.y * a.y;

#pragma unroll
    for (int off = 16; off; off >>= 1) p += __shfl_xor(p, off, 32);

    if (lane == 0) {
        logits[e] = p;
        atomicMax(lmax + d, fkey(p));
    }
}

// One thread per edge: ex = exp(logit - max[dst]); denom[dst] += ex.
// In-place: exv may alias logits (same-index read-then-write).
__global__ void edge_exp(const long long* __restrict__ ei,
                         const float* __restrict__ logits,
                         const unsigned* __restrict__ lmax,
                         float* __restrict__ exv, float* __restrict__ denom, int E)
{
    const int e = blockIdx.x * blockDim.x + threadIdx.x;
    if (e >= E) return;
    const int d = (int)ei[(size_t)E + e];
    const float v = __expf(logits[e] - funkey(lmax[d]));
    exv[e] = v;
    atomicAdd(denom + d, v);
}

// One wave per edge: out[dst] += (ex/denom[dst]) * xl[src], f32 atomics
// coalesced across the 256B destination row.
__global__ __launch_bounds__(256) void edge_scatter(
    const long long* __restrict__ ei, const float* __restrict__ exv,
    const float* __restrict__ denom, const float* __restrict__ xl,
    float* __restrict__ out, int E)
{
    const int e = blockIdx.x * (blockDim.x >> 5) + (threadIdx.x >> 5);
    if (e >= E) return;                               // wave-uniform
    const int lane = threadIdx.x & 31;

    const int s = (int)ei[e];
    const int d = (int)ei[(size_t)E + e];
    const float alpha = exv[e] / (denom[d] + 1e-16f);

    const v2f* xl2 = (const v2f*)(xl + (size_t)s * CH);
    const v2f v = alpha * xl2[lane];
    float* o = out + (size_t)d * CH + lane * 2;
    atomicAdd(o, v.x);
    atomicAdd(o + 1, v.y);
}

extern "C" void kernel_launch(void* const* d_in, const int* in_sizes, int n_in,
                              void* d_out, int out_size, void* d_ws, size_t ws_size,
                              hipStream_t stream)
{
    const float*     x  = (const float*)d_in[0];
    const long long* ei = (const long long*)d_in[1];   // int64 edge_index [2,E]
    const float*     ea = (const float*)d_in[2];

    const int DIN = 128;
    const int N = in_sizes[0] / DIN;                   // 50000
    const int E = in_sizes[2];                         // 1600000

    // Workspace layout
    char* ws = (char*)d_ws;
    float*    xl    = (float*)ws;             ws += (size_t)N * CH * sizeof(float);
    float*    xr    = (float*)ws;             ws += (size_t)N * CH * sizeof(float);
    float*    hA    = (float*)ws;             ws += (size_t)N * CH * sizeof(float);
    float*    hB    = (float*)ws;             ws += (size_t)N * CH * sizeof(float);
    float*    elog  = (float*)ws;             ws += (size_t)E * sizeof(float);      // logits -> ex (in place)
    unsigned* lmax  = (unsigned*)ws;          ws += (size_t)N * sizeof(unsigned);
    float*    denom = (float*)ws;             ws += (size_t)N * sizeof(float);

    const float* layer_in[3]  = { x,  hA, hB };
    float*       layer_out[3] = { hA, hB, (float*)d_out };
    const int    layer_K[3]   = { DIN, CH, CH };

    const int tilesM = (N + 15) / 16;
    const int tilesN = CH / 16;
    const int gemmBlocks  = (tilesM * tilesN + 7) / 8;     // 8 waves / 256-thread block
    const int initBlocks  = (N * CH + 255) / 256;
    const int edgeWBlocks = (E + 7) / 8;                   // one wave per edge
    const int edgeTBlocks = (E + 255) / 256;

    for (int l = 0; l < 3; ++l) {
        const int base = 3 + 7 * l;
        const float* Wl  = (const float*)d_in[base + 0];
        const float* bl  = (const float*)d_in[base + 1];
        const float* Wr  = (const float*)d_in[base + 2];
        const float* br  = (const float*)d_in[base + 3];
        const float* We  = (const float*)d_in[base + 4];
        const float* att = (const float*)d_in[base + 5];
        const float* b   = (const float*)d_in[base + 6];

        init_layer<<<initBlocks, 256, 0, stream>>>(layer_out[l], b, lmax, denom, N);

        if (l == 0) {
            gemm_wmma_f32<0><<<gemmBlocks, 256, 0, stream>>>(layer_in[l], Wl, bl, xl,
                                                             N, layer_K[l]);
            gemm_wmma_f32<0><<<gemmBlocks, 256, 0, stream>>>(layer_in[l], Wr, br, xr,
                                                             N, layer_K[l]);
        } else {
            // ReLU on previous layer's output, fused into the A-fragment load.
            gemm_wmma_f32<1><<<gemmBlocks, 256, 0, stream>>>(layer_in[l], Wl, bl, xl,
                                                             N, layer_K[l]);
            gemm_wmma_f32<1><<<gemmBlocks, 256, 0, stream>>>(layer_in[l], Wr, br, xr,
                                                             N, layer_K[l]);
        }

        edge_logits<<<edgeWBlocks, 256, 0, stream>>>(ei, ea, xl, xr, We, att,
                                                     elog, lmax, E);
        edge_exp<<<edgeTBlocks, 256, 0, stream>>>(ei, elog, lmax, elog, denom, E);
        edge_scatter<<<edgeWBlocks, 256, 0, stream>>>(ei, elog, denom, xl,
                                                      layer_out[l], E);
    }
}